// involution_463856468144
// MI455X (gfx1250) — compile-verified
//
#include <hip/hip_runtime.h>

// Involution (B=32, C=1, H=W=256, K=7, C_hid=16) for gfx1250.
//
// h_c = relu(a_c*x + d_c) (w1/b1/BN folded into a,d). The dominant matvec
// weight = W2(49x16) @ h is run on V_WMMA_F32_16X16X4_F32: W2 zero-padded to
// 64x16, 4 row-tiles x 4 K-steps = 16 fp32 WMMAs per 16-pixel wave tile.
// b2 is pre-folded into the WMMA C operand. Tap application uses three
// per-lane LDS base pointers so every tap is ds_load_b32(imm) + v_fmac,
// with all addresses statically in-bounds of the initialized s_x halo.

typedef __attribute__((ext_vector_type(2))) float v2f;
typedef __attribute__((ext_vector_type(4))) float v4f;
typedef __attribute__((ext_vector_type(8))) float v8f;

#define KS     7
#define PADR   3
#define CH     16
#define KK     49                  // 7*7 taps
#define KKP    64                  // padded rows for WMMA
#define TILE_W 128
#define HALO_W (TILE_W + KS - 1)   // 134
#define NX     (KS * HALO_W)       // 938 floats of halo
#define BN_EPS 1e-5f

__global__ __launch_bounds__(256) void involution_wmma_kernel(
    const float* __restrict__ x,
    const float* __restrict__ w1,  const float* __restrict__ b1,
    const float* __restrict__ gam, const float* __restrict__ bet,
    const float* __restrict__ mu,  const float* __restrict__ var,
    const float* __restrict__ w2,  const float* __restrict__ b2,
    float* __restrict__ out, int B, int H, int W)
{
    __shared__ __align__(16) float s_x[NX];         // halo strip (first!)
    __shared__ __align__(16) float s_w2[KKP][CH];   // zero-padded rows 49..63
    __shared__ __align__(32) float s_b2[KKP];       // zero-padded
    __shared__ __align__(16) float s_ad[2 * CH];    // interleaved (a_c, d_c)

    const int tid = threadIdx.x;

    for (int idx = tid; idx < KKP * CH; idx += 256) {
        const int k = idx >> 4, c = idx & 15;
        s_w2[k][c] = (k < KK) ? w2[k * CH + c] : 0.f;
    }
    if (tid < KKP) s_b2[tid] = (tid < KK) ? b2[tid] : 0.f;
    if (tid < CH) {
        const float inv = gam[tid] * rsqrtf(var[tid] + BN_EPS);
        s_ad[2 * tid]     = w1[tid] * inv;
        s_ad[2 * tid + 1] = (b1[tid] - mu[tid]) * inv + bet[tid];
    }

    const int col0 = blockIdx.x * TILE_W;
    const int row  = blockIdx.y;
    const int b    = blockIdx.z;
    const float* __restrict__ xb = x + (size_t)b * H * W;

    for (int idx = tid; idx < NX; idx += 256) {
        const int r  = idx / HALO_W;
        const int c  = idx - r * HALO_W;
        const int gi = row - PADR + r;
        const int gj = col0 - PADR + c;
        float v = 0.f;
        if (gi >= 0 && gi < H && gj >= 0 && gj < W) v = xb[gi * W + gj];
        s_x[idx] = v;
    }
    __syncthreads();

    const int lane = tid & 31;
    const int wave = tid >> 5;
    const int half = lane >> 4;   // K/row half this lane holds in the WMMA
    const int p    = lane & 15;   // pixel (N column) within the wave tile
    const int j    = wave * 16 + p;

    const float xc = s_x[PADR * HALO_W + PADR + j];

    // ---- A operands: W2 row = p, K pair = 4s + 2*half (8B-aligned v2f) ----
    v2f A[4][4];
#pragma unroll
    for (int t = 0; t < 4; ++t)
#pragma unroll
        for (int s = 0; s < 4; ++s)
            A[t][s] = *(const v2f*)&s_w2[t * 16 + p][s * 4 + 2 * half];

    // ---- C init = b2 rows owned by this lane (contiguous b128 loads) ----
    const float* pb2 = &s_b2[8 * half];
    v8f acc[4];
#pragma unroll
    for (int t = 0; t < 4; ++t)
        acc[t] = *(const v8f*)&pb2[16 * t];

    // ---- weight(64 x 16px) = W2pad @ h + b2 : 16x fp32 WMMAs ----
    const float* pada = &s_ad[4 * half];   // (a,d,a,d) for this lane's K pair
#pragma unroll
    for (int s = 0; s < 4; ++s) {
        const v4f ad = *(const v4f*)&pada[8 * s];   // byte off 32s+16half
        v2f Bv;
        Bv.x = fmaxf(fmaf(ad.x, xc, ad.y), 0.f);
        Bv.y = fmaxf(fmaf(ad.z, xc, ad.w), 0.f);
#pragma unroll
        for (int t = 0; t < 4; ++t) {
            acc[t] = __builtin_amdgcn_wmma_f32_16x16x4_f32(
                false, A[t][s], false, Bv, (short)0, acc[t], false, false);
        }
    }

    // ---- apply dynamic 7x7 kernel ----
    // Lane owns rows m = 16t + v + 8*half. Relative to the half=0 tap offset,
    // half=1 adds +135 (m+8 = +1 row +1 col), +262 when mlo%7==6 (col wrap),
    // and 0 for rows with mhi>=49 (their wgt is exactly 0; address clamped
    // in-bounds so no uninitialized LDS is ever multiplied in).
    const float* q0 = &s_x[j];
    const float* q1 = q0 + 135 * half;
    const float* q2 = q0 + 262 * half;

    float partial = 0.f;
#pragma unroll
    for (int t = 0; t < 4; ++t) {
#pragma unroll
        for (int v = 0; v < 8; ++v) {
            const int mlo  = 16 * t + v;
            const int mhi  = mlo + 8;
            const int mloc = (mlo < KK) ? mlo : 48;              // clamp
            const int off  = (mloc / KS) * HALO_W + (mloc % KS); // <= 810
            const float* q = (mhi >= KK) ? q0 : ((mlo % KS == 6) ? q2 : q1);
            partial = fmaf(acc[t][v], q[off], partial);
        }
    }

    // combine the two lane-halves covering the same pixel
    partial += __shfl_xor(partial, 16, 32);

    if (half == 0)
        out[((size_t)b * H + row) * W + col0 + j] = partial;
}

extern "C" void kernel_launch(void* const* d_in, const int* in_sizes, int n_in,
                              void* d_out, int out_size, void* d_ws, size_t ws_size,
                              hipStream_t stream) {
    (void)n_in; (void)d_ws; (void)ws_size; (void)out_size;
    const float* x   = (const float*)d_in[0];
    const float* w1  = (const float*)d_in[1];
    const float* b1  = (const float*)d_in[2];
    const float* gam = (const float*)d_in[3];
    const float* bet = (const float*)d_in[4];
    const float* mu  = (const float*)d_in[5];
    const float* var = (const float*)d_in[6];
    const float* w2  = (const float*)d_in[7];
    const float* b2  = (const float*)d_in[8];
    float* out = (float*)d_out;

    const int H = 256, W = 256;
    const int B = in_sizes[0] / (H * W);   // 32

    dim3 grid(W / TILE_W, H, B);           // (2, 256, 32)
    dim3 block(256);
    involution_wmma_kernel<<<grid, block, 0, stream>>>(
        x, w1, b1, gam, bet, mu, var, w2, b2, out, B, H, W);
}